// PEG_Shift_Q_41480794144827
// MI455X (gfx1250) — compile-verified
//
#include <hip/hip_runtime.h>

// Depthwise 3x3 (stride 1, pad 1) with DeepShift quantization.
// x: (32, 768, 56, 56) f32, w: (768,1,3,3) f32, bias: (768,) f32 -> out same shape as x.
// Bandwidth-bound: ~616 MB @ 23.3 TB/s => ~26.5 us floor. WMMA is inapplicable
// (K=9 per-channel reduction); use CDNA5 async global->LDS DMA instead.

#define CHANNELS 768
#define HW       56
#define PLANE    (HW * HW)        // 3136 floats per (n,c) plane
#define TILE_W   60               // padded tile row stride (keeps 16B row alignment)
#define TILE_H   58               // 56 + halo
#define NCHUNK   (PLANE / 4)      // 784 x 16B chunks per plane
#define TPB      256

typedef __attribute__((ext_vector_type(4))) float v4f;

__device__ __forceinline__ float quant_act(float x) {
    // floor(x / 2^-16 + 0.5) * 2^-16, clipped to [-2^15, 2^15 - 1]
    float t = x * 65536.0f;           // exact same op order as reference
    t = floorf(t + 0.5f);
    t = t * (1.0f / 65536.0f);
    t = fmaxf(t, -32768.0f);
    t = fminf(t,  32767.0f);
    return t;
}

__device__ __forceinline__ float quant_w(float w) {
    // sign(w) * 2^clip(round(log2(|w| + 1e-32)), -14, 0); sign(0) = 0
    float a  = fabsf(w) + 1e-32f;
    float sh = rintf(log2f(a));       // rint = round-half-even, matches jnp.round
    sh = fminf(fmaxf(sh, -14.0f), 0.0f);
    float m = ldexpf(1.0f, (int)sh);  // exact power of two
    return (w > 0.0f) ? m : ((w < 0.0f) ? -m : 0.0f);
}

__global__ __launch_bounds__(TPB) void dwconv3x3_shift_kernel(
    const float* __restrict__ x,
    const float* __restrict__ wgt,
    const float* __restrict__ bias,
    float*       __restrict__ out)
{
    __shared__ __align__(16) float rawA[PLANE];            // 12544 B: raw plane (DMA target)
    __shared__ __align__(16) float tileB[TILE_H * TILE_W]; // 13920 B: quantized, zero-padded

    const int tid   = threadIdx.x;
    const int plane = blockIdx.x;        // n * 768 + c
    const int c     = plane % CHANNELS;
    const float* xp = x   + (size_t)plane * PLANE;
    float*       op = out + (size_t)plane * PLANE;

    // ---- Phase 0: async DMA global -> LDS (no VGPR staging), 16B per lane-chunk.
    // All global and LDS addresses are 16B-aligned (plane = 12544 B, rawA aligned 16).
    {
        unsigned ldsA = (unsigned)(uintptr_t)rawA;   // low 32 bits of generic ptr = LDS offset
        unsigned long long gbase = (unsigned long long)(uintptr_t)xp;
        for (int k = tid; k < NCHUNK; k += TPB) {
            unsigned la = ldsA + (unsigned)k * 16u;
            unsigned long long ga = gbase + (unsigned long long)k * 16u;
            asm volatile("global_load_async_to_lds_b128 %0, %1, off"
                         :: "v"(la), "v"(ga)
                         : "memory");
        }
    }

    // ---- Phase 1: zero the padded tile while the DMA is in flight (halo = conv zero-pad).
    {
        v4f z = {0.0f, 0.0f, 0.0f, 0.0f};
        v4f* b4 = (v4f*)tileB;
        for (int k = tid; k < (TILE_H * TILE_W) / 4; k += TPB) b4[k] = z;
    }

    // Quantized 3x3 weights + bias (uniform per block -> scalar loads).
    float wq[9];
    #pragma unroll
    for (int j = 0; j < 9; ++j) wq[j] = quant_w(wgt[c * 9 + j]);
    const float bv = bias[c];

#if __has_builtin(__builtin_amdgcn_s_wait_asynccnt)
    __builtin_amdgcn_s_wait_asynccnt(0);
#else
    asm volatile("s_wait_asynccnt 0x0" ::: "memory");
#endif
    __syncthreads();

    // ---- Phase 2: fixed-point-quantize each input element ONCE into the padded tile.
    for (int k = tid; k < NCHUNK; k += TPB) {
        const unsigned uk = (unsigned)k;
        const int h  = (int)(uk / 14u);
        const int w0 = (int)(uk % 14u) * 4;
        v4f v = *(const v4f*)(rawA + k * 4);
        float* dst = tileB + (h + 1) * TILE_W + (w0 + 1);
        dst[0] = quant_act(v.x);
        dst[1] = quant_act(v.y);
        dst[2] = quant_act(v.z);
        dst[3] = quant_act(v.w);
    }
    __syncthreads();

    // ---- Phase 3: branchless 3x3 depthwise conv, 4 outputs per thread-strip.
    for (int k = tid; k < NCHUNK; k += TPB) {
        const unsigned uk = (unsigned)k;
        const int h  = (int)(uk / 14u);
        const int w0 = (int)(uk % 14u) * 4;
        float a0 = bv, a1 = bv, a2 = bv, a3 = bv;
        #pragma unroll
        for (int r = 0; r < 3; ++r) {
            const float* row = tileB + (h + r) * TILE_W + w0;
            const float f0 = row[0], f1 = row[1], f2 = row[2];
            const float f3 = row[3], f4 = row[4], f5 = row[5];
            const float k0 = wq[3 * r], k1 = wq[3 * r + 1], k2 = wq[3 * r + 2];
            a0 = fmaf(f2, k2, fmaf(f1, k1, fmaf(f0, k0, a0)));
            a1 = fmaf(f3, k2, fmaf(f2, k1, fmaf(f1, k0, a1)));
            a2 = fmaf(f4, k2, fmaf(f3, k1, fmaf(f2, k0, a2)));
            a3 = fmaf(f5, k2, fmaf(f4, k1, fmaf(f3, k0, a3)));
        }
        v4f o = {a0, a1, a2, a3};
        // Output is write-once, never re-read by this kernel: nontemporal b128 store.
        __builtin_nontemporal_store(o, (v4f*)(op + h * HW + w0));
    }
}

extern "C" void kernel_launch(void* const* d_in, const int* in_sizes, int n_in,
                              void* d_out, int out_size, void* d_ws, size_t ws_size,
                              hipStream_t stream) {
    const float* x = (const float*)d_in[0];
    const float* w = (const float*)d_in[1];
    const float* b = (const float*)d_in[2];
    float* out = (float*)d_out;
    const int nplanes = in_sizes[0] / PLANE;   // 32 * 768 = 24576
    dwconv3x3_shift_kernel<<<nplanes, TPB, 0, stream>>>(x, w, b, out);
}